// GlobalAttentionGeneral_76871324664031
// MI455X (gfx1250) — compile-verified
//
#include <hip/hip_runtime.h>
#include <hip/hip_bf16.h>
#include <stddef.h>

// Problem constants (from reference setup_inputs)
#define BB   32
#define IDF_ 512
#define CDF_ 512
#define QL   4096   // 64*64
#define SL   300
#define PP   16
#define NP   32     // padded p-dimension (17 valid rows -> pad to 32)

typedef __attribute__((ext_vector_type(2))) float v2f;
typedef __attribute__((ext_vector_type(8))) float v8f;
typedef __attribute__((ext_vector_type(4))) float v4f;
typedef __attribute__((ext_vector_type(4))) int   v4i;

// ---------------------------------------------------------------------------
// Kernel P: build p2aT (SL x 32, padded with zeros) and rowsum[p] = sum_s p2a
// ---------------------------------------------------------------------------
__global__ void prep_kernel(const int* __restrict__ part2attr,
                            float* __restrict__ p2aT,
                            float* __restrict__ rowsum) {
  const int t = threadIdx.x;  // one block of 256
  for (int idx = t; idx < SL * NP; idx += 256) {
    const int s = idx >> 5;
    const int p = idx & (NP - 1);
    p2aT[idx] = (p < PP + 1) ? (float)part2attr[p * SL + s] : 0.0f;
  }
  if (t < NP) {
    float acc = 0.0f;
    if (t < PP + 1) {
      for (int s = 0; s < SL; ++s) acc += (float)part2attr[t * SL + s];
    }
    rowsum[t] = acc;
  }
}

// ---------------------------------------------------------------------------
// Kernel A: max_seg[b,q] = max over i of (i+1 if seg[b,i,q]!=0 else 0)
// ---------------------------------------------------------------------------
__global__ void maxseg_kernel(const int* __restrict__ seg,
                              int* __restrict__ maxseg) {
  const int idx = blockIdx.x * blockDim.x + threadIdx.x;  // 0 .. B*QL-1
  const int b = idx >> 12;
  const int q = idx & (QL - 1);
  const int* s = seg + (size_t)b * PP * QL + q;
  int m = 0;
#pragma unroll
  for (int i = 0; i < PP; ++i) {
    if (s[(size_t)i * QL] != 0) m = i + 1;
  }
  maxseg[idx] = m;
}

// ---------------------------------------------------------------------------
// Kernel C: ctxAgg[b, c, p] = sum_s context[b,c,s] * p2aT[s,p]
//   per wave: one 16x32 tile (M = 16 channels, N = 32 padded p), K = SL = 300
//   uses V_WMMA_F32_16X16X4_F32 (fp32 in/out), 75 k-steps, 2 wmma per step
// ---------------------------------------------------------------------------
__global__ void __launch_bounds__(32)
ctxagg_kernel(const float* __restrict__ context,
              const float* __restrict__ p2aT,
              float* __restrict__ ctxAgg) {
  const int mt   = blockIdx.x;       // 0..31 channel tile
  const int b    = blockIdx.y;       // batch
  const int lane = threadIdx.x;      // 0..31 (wave32)
  const int row   = lane & 15;       // M index within tile
  const int khalf = (lane >> 4) * 2; // 0 or 2  (A/B K-split per ISA layout)
  const int col   = lane & 15;       // N index within 16-wide tile

  const float* A = context + (size_t)b * CDF_ * SL + (size_t)(mt * 16 + row) * SL;

  v8f c0 = {};
  v8f c1 = {};
  for (int k = 0; k < SL; k += 4) {
    v2f a;
    a.x = A[k + khalf];
    a.y = A[k + khalf + 1];
    v2f b0, b1;
    const float* Br0 = p2aT + (size_t)(k + khalf) * NP;
    const float* Br1 = Br0 + NP;
    b0.x = Br0[col];       b0.y = Br1[col];
    b1.x = Br0[col + 16];  b1.y = Br1[col + 16];
    c0 = __builtin_amdgcn_wmma_f32_16x16x4_f32(false, a, false, b0, (short)0, c0, false, false);
    c1 = __builtin_amdgcn_wmma_f32_16x16x4_f32(false, a, false, b1, (short)0, c1, false, false);
  }

  // D layout: VGPR r holds M = r (lanes 0-15) / M = r+8 (lanes 16-31), N = lane&15
  float* out = ctxAgg + (size_t)b * CDF_ * NP + (size_t)(mt * 16) * NP;
  const int half = lane >> 4;
#pragma unroll
  for (int r = 0; r < 8; ++r) {
    const int rr = r + 8 * half;
    out[rr * NP + col]      = c0[r];
    out[rr * NP + col + 16] = c1[r];
  }
}

// ---------------------------------------------------------------------------
// Kernel D: agg[b, d, p] = sum_c Wc[d,c] * ctxAgg[b,c,p] + bc[d] * rowsum[p]
//   per wave: one 16x32 tile, K = 512 (128 k-steps of 4)
// ---------------------------------------------------------------------------
__global__ void __launch_bounds__(32)
agg_kernel(const float* __restrict__ Wc,
           const float* __restrict__ bc,
           const float* __restrict__ rowsum,
           const float* __restrict__ ctxAgg,
           float* __restrict__ agg) {
  const int mt   = blockIdx.x;       // 0..31 d tile
  const int b    = blockIdx.y;       // batch
  const int lane = threadIdx.x;      // 0..31
  const int row   = lane & 15;
  const int khalf = (lane >> 4) * 2;
  const int col   = lane & 15;

  const float* A  = Wc + (size_t)(mt * 16 + row) * CDF_;
  const float* Bm = ctxAgg + (size_t)b * CDF_ * NP;

  v8f c0 = {};
  v8f c1 = {};
  for (int k = 0; k < CDF_; k += 4) {
    v2f a;
    a.x = A[k + khalf];
    a.y = A[k + khalf + 1];
    v2f b0, b1;
    const float* Br0 = Bm + (size_t)(k + khalf) * NP;
    const float* Br1 = Br0 + NP;
    b0.x = Br0[col];       b0.y = Br1[col];
    b1.x = Br0[col + 16];  b1.y = Br1[col + 16];
    c0 = __builtin_amdgcn_wmma_f32_16x16x4_f32(false, a, false, b0, (short)0, c0, false, false);
    c1 = __builtin_amdgcn_wmma_f32_16x16x4_f32(false, a, false, b1, (short)0, c1, false, false);
  }

  float* out = agg + (size_t)b * IDF_ * NP + (size_t)(mt * 16) * NP;
  const int half = lane >> 4;
#pragma unroll
  for (int r = 0; r < 8; ++r) {
    const int rr   = r + 8 * half;
    const int drow = mt * 16 + rr;
    const float bias = bc[drow];
    out[rr * NP + col]      = c0[r] + bias * rowsum[col];
    out[rr * NP + col + 16] = c1[r] + bias * rowsum[col + 16];
  }
}

// ---------------------------------------------------------------------------
// Kernel E: weightedContext[b,d,q] = agg[b, d, max_seg[b,q]]   (268 MB stream)
//   vectorized: v4i max_seg loads + v4f NT stores (512 B / wave / store)
// ---------------------------------------------------------------------------
__global__ void wc_gather_kernel(const float* __restrict__ agg,
                                 const int* __restrict__ maxseg,
                                 float* __restrict__ out) {
  __shared__ float lut[PP + 1];
  const int d = blockIdx.x;
  const int b = blockIdx.y;
  if (threadIdx.x < PP + 1)
    lut[threadIdx.x] = agg[((size_t)b * IDF_ + d) * NP + threadIdx.x];
  __syncthreads();
  const v4i* ms4 = (const v4i*)(maxseg + (size_t)b * QL);
  v4f* o4 = (v4f*)(out + ((size_t)b * IDF_ + d) * QL);
  for (int q = threadIdx.x; q < QL / 4; q += blockDim.x) {
    const v4i m = ms4[q];
    v4f v;
    v.x = lut[m.x]; v.y = lut[m.y]; v.z = lut[m.z]; v.w = lut[m.w];
    __builtin_nontemporal_store(v, &o4[q]);
  }
}

// ---------------------------------------------------------------------------
// Kernel F: attn_out[b,s,q] = (float)part2attr[max_seg[b,q], s]  (157 MB stream)
// ---------------------------------------------------------------------------
__global__ void attn_gather_kernel(const int* __restrict__ part2attr,
                                   const int* __restrict__ maxseg,
                                   float* __restrict__ out) {
  __shared__ float lut[PP + 1];
  const int s = blockIdx.x;
  const int b = blockIdx.y;
  if (threadIdx.x < PP + 1)
    lut[threadIdx.x] = (float)part2attr[threadIdx.x * SL + s];
  __syncthreads();
  const v4i* ms4 = (const v4i*)(maxseg + (size_t)b * QL);
  v4f* o4 = (v4f*)(out + ((size_t)b * SL + s) * QL);
  for (int q = threadIdx.x; q < QL / 4; q += blockDim.x) {
    const v4i m = ms4[q];
    v4f v;
    v.x = lut[m.x]; v.y = lut[m.y]; v.z = lut[m.z]; v.w = lut[m.w];
    __builtin_nontemporal_store(v, &o4[q]);
  }
}

// ---------------------------------------------------------------------------
extern "C" void kernel_launch(void* const* d_in, const int* in_sizes, int n_in,
                              void* d_out, int out_size, void* d_ws, size_t ws_size,
                              hipStream_t stream) {
  (void)in_sizes; (void)n_in; (void)out_size; (void)ws_size;
  // inputs: 0=input (unused, shape only), 1=context, 2=seg, 3=part2attr, 4=Wc, 5=bc
  const float* context   = (const float*)d_in[1];
  const int*   seg       = (const int*)d_in[2];
  const int*   part2attr = (const int*)d_in[3];
  const float* Wc        = (const float*)d_in[4];
  const float* bc        = (const float*)d_in[5];

  // workspace layout (floats): ~4.8 MB total
  int*   maxseg = (int*)d_ws;                         // B*QL           = 131072
  float* p2aT   = (float*)d_ws + (BB * QL);           // SL*NP          = 9600
  float* rowsum = p2aT + SL * NP;                     // NP             = 32
  float* ctxAgg = rowsum + NP;                        // B*CDF*NP       = 524288
  float* agg    = ctxAgg + (size_t)BB * CDF_ * NP;    // B*IDF*NP       = 524288

  float* wc_out   = (float*)d_out;                    // (B, 512, 64, 64)
  float* attn_out = wc_out + (size_t)BB * IDF_ * QL;  // (B, 300, 64, 64)

  // 1) tiny prep: transpose/pad LUT + row sums
  prep_kernel<<<1, 256, 0, stream>>>(part2attr, p2aT, rowsum);

  // 2) max_seg over 16 planes
  maxseg_kernel<<<(BB * QL) / 256, 256, 0, stream>>>(seg, maxseg);

  // 3) ctxAgg = context @ p2aT   (WMMA f32, one wave per 16x32 tile)
  ctxagg_kernel<<<dim3(CDF_ / 16, BB), 32, 0, stream>>>(context, p2aT, ctxAgg);

  // 4) agg = Wc @ ctxAgg + bc*rowsum   (WMMA f32)
  agg_kernel<<<dim3(IDF_ / 16, BB), 32, 0, stream>>>(Wc, bc, rowsum, ctxAgg, agg);

  // 5) streamed gathers (memory-bound: ~425 MB of vectorized NT stores)
  wc_gather_kernel<<<dim3(IDF_, BB), 256, 0, stream>>>(agg, maxseg, wc_out);
  attn_gather_kernel<<<dim3(SL, BB), 256, 0, stream>>>(part2attr, maxseg, attn_out);
}